// ConceptNet_58514634441291
// MI455X (gfx1250) — compile-verified
//
#include <hip/hip_runtime.h>
#include <hip/hip_bf16.h>
#include <float.h>

// Problem sizes (fixed by the reference)
#define DDIM 1024
#define NTE  200000
#define CCON 50
#define BB   4096
#define NCLS 4
#define MPAD 64
#define LDSS 68  // LDS row stride (dwords): 16B-aligned rows, conflict-free frags

typedef float v2f __attribute__((ext_vector_type(2)));
typedef float v8f __attribute__((ext_vector_type(8)));

#if __has_builtin(__builtin_amdgcn_global_load_async_to_lds_b128)
#define HAVE_ASYNC_LDS 1
// exact param types per hipcc diagnostic: int vector_size(16), AS-qualified
typedef int avi4 __attribute__((vector_size(16)));
typedef __attribute__((address_space(1))) avi4 gvi4;
typedef __attribute__((address_space(3))) avi4 lvi4;
#endif

__device__ __forceinline__ void wait_async0() {
#if __has_builtin(__builtin_amdgcn_s_wait_asynccnt)
  __builtin_amdgcn_s_wait_asynccnt(0);
#else
  asm volatile("s_wait_asynccnt 0" ::: "memory");
#endif
}

// ---- workspace layout (floats) ----
#define OFF_DOTS 0ull                                  // 50 * 200000
#define OFF_TESQ (OFF_DOTS + (size_t)CCON * NTE)       // 200000
#define OFF_XT   (OFF_TESQ + (size_t)NTE)              // 1024*4096
#define OFF_T1   (OFF_XT + (size_t)DDIM * BB)          // 50*4096
#define OFF_GRAM (OFF_T1 + (size_t)CCON * BB)          // 2500
#define OFF_INVG (OFF_GRAM + (size_t)CCON * CCON)      // 2500
#define OFF_W3   (OFF_INVG + (size_t)CCON * CCON)      // 200
#define OFF_CSUM (OFF_W3 + (size_t)NCLS * CCON)        // 64
#define OFF_APAD (OFF_CSUM + 64ull)                    // 64*1024 (K-major)

// ---------------------------------------------------------------------------
// ApadT[m*1024 + d] = (m<50) ? concept[d*50+m] : 0   (K-major padded A)
// ---------------------------------------------------------------------------
__global__ __launch_bounds__(256) void padA_kernel(const float* __restrict__ con,
                                                   float* __restrict__ apadt) {
  const int idx = blockIdx.x * 256 + threadIdx.x;
  if (idx >= MPAD * DDIM) return;
  const int m = idx >> 10, d = idx & 1023;
  apadt[idx] = (m < CCON) ? con[d * CCON + m] : 0.f;
}

// ---------------------------------------------------------------------------
// XT[d*BB + b] = X[b*DDIM + d]  (32x32 LDS tile transpose)
// ---------------------------------------------------------------------------
__global__ __launch_bounds__(256) void transpose_kernel(const float* __restrict__ X,
                                                        float* __restrict__ XT) {
  __shared__ float t[32][33];
  const int dt = blockIdx.x * 32, bt = blockIdx.y * 32;
  const int tx = threadIdx.x, ty = threadIdx.y;
#pragma unroll
  for (int i = 0; i < 32; i += 8)
    t[ty + i][tx] = X[(size_t)(bt + ty + i) * DDIM + dt + tx];
  __syncthreads();
#pragma unroll
  for (int i = 0; i < 32; i += 8)
    XT[(size_t)(dt + ty + i) * BB + bt + tx] = t[tx][ty + i];
}

// ---------------------------------------------------------------------------
// gram[i,j] = sum_d concept[d*CCON+i] * concept[d*CCON+j]
// ---------------------------------------------------------------------------
__global__ void gram_kernel(const float* __restrict__ con, float* __restrict__ gram) {
  const int idx = blockIdx.x * blockDim.x + threadIdx.x;
  if (idx >= CCON * CCON) return;
  const int i = idx / CCON, j = idx % CCON;
  float s = 0.f;
  for (int d = 0; d < DDIM; ++d)
    s = fmaf(con[d * CCON + i], con[d * CCON + j], s);
  gram[idx] = s;
}

// ---------------------------------------------------------------------------
// Gauss-Jordan inverse of the (SPD) 50x50 gram matrix, single block, LDS.
// ---------------------------------------------------------------------------
__global__ __launch_bounds__(64) void inv_kernel(const float* __restrict__ gram,
                                                 float* __restrict__ invg) {
  __shared__ float aug[CCON][2 * CCON + 2];
  const int tid = threadIdx.x;
  if (tid < CCON) {
    for (int j = 0; j < CCON; ++j) aug[tid][j] = gram[tid * CCON + j];
    for (int j = 0; j < CCON; ++j) aug[tid][CCON + j] = (tid == j) ? 1.f : 0.f;
  }
  __syncthreads();
  for (int p = 0; p < CCON; ++p) {
    if (tid == p) {
      const float inv = 1.0f / aug[p][p];
      for (int j = 0; j < 2 * CCON; ++j) aug[p][j] *= inv;
    }
    __syncthreads();
    if (tid < CCON && tid != p) {
      const float f = aug[tid][p];
      for (int j = 0; j < 2 * CCON; ++j)
        aug[tid][j] = fmaf(-f, aug[p][j], aug[tid][j]);
    }
    __syncthreads();
  }
  if (tid < CCON)
    for (int j = 0; j < CCON; ++j) invg[tid * CCON + j] = aug[tid][CCON + j];
}

// ---------------------------------------------------------------------------
// WMMA f32 skinny GEMM:  OUT[m,n] = sum_d ApadT[m*1024+d] * Bmat[d*ncols+n]
// M = 64 (4 x 16 tiles), 8 waves x 16 cols = 128 cols per workgroup.
// A chunks staged K-major via GLOBAL_LOAD_ASYNC_TO_LDS_B128; fragment fetch is
// one conflict-free ds_load_b64 per m-tile (pairs land in aligned VGPRs).
// B streamed global->VGPR (single-use), one-step software pipeline,
// fused te_sq via lane sums + shfl_xor(16).
// A-fragment (16x4 f32): lanes 0-15 hold K=k,k+1; lanes 16-31 hold K=k+2,k+3.
// C/D (16x16 f32): VGPR r -> M=r (lanes 0-15) / M=r+8 (lanes 16-31).
// ---------------------------------------------------------------------------
__global__ __launch_bounds__(256) void wmma_gemm_kernel(
    const float* __restrict__ ApadT, const float* __restrict__ Bmat,
    float* __restrict__ outRows, float* __restrict__ tesq, int ncols) {
  __shared__ __align__(16) float ldsA[MPAD * LDSS];  // [m][k_local], ~17 KB
  const int tid = threadIdx.x;
  const int wave = tid >> 5;
  const int lane = tid & 31;
  const int half = lane >> 4;  // 0: rows k,k+1   1: rows k+2,k+3
  const int ml = lane & 15;
  const int col = blockIdx.x * 128 + wave * 16 + ml;
  const bool cok = col < ncols;
  const int colc = cok ? col : (ncols - 1);  // clamped: safe addr, result discarded
  const size_t rs = (size_t)ncols;

  v8f acc[4];
#pragma unroll
  for (int t = 0; t < 4; ++t)
#pragma unroll
    for (int r = 0; r < 8; ++r) acc[t][r] = 0.f;
  float ss = 0.f;

  // software-pipelined B pointer: bq addresses the pair currently in (b0,b1)
  const float* bq = Bmat + (size_t)(2 * half) * rs + colc;
  float b0 = bq[0];
  float b1 = bq[rs];
  int knext = 4;  // uniform next-k row counter

  for (int k0 = 0; k0 < DDIM; k0 += 64) {
    // ---- stage ApadT[0..63][k0..k0+63] into LDS (async copy, 16B units) ----
#ifdef HAVE_ASYNC_LDS
    for (int e = tid; e < 1024; e += 256) {
      const int m = e >> 4, q = e & 15;
      __builtin_amdgcn_global_load_async_to_lds_b128(
          (gvi4*)(ApadT + (size_t)m * DDIM + k0 + q * 4),
          (lvi4*)(ldsA + m * LDSS + q * 4), 0, 0);
    }
    wait_async0();
#else
    for (int e = tid; e < 1024; e += 256) {
      const int m = e >> 4, q = e & 15;
      *(float4*)(ldsA + m * LDSS + q * 4) =
          *(const float4*)(ApadT + (size_t)m * DDIM + k0 + q * 4);
    }
#endif
    __syncthreads();

    if (k0 + 64 < DDIM) __builtin_prefetch(bq + 64 * rs, 0, 1);

#pragma unroll 4
    for (int kk = 0; kk < 64; kk += 4) {
      // prefetch next B pair into registers (wraps to a safe address at the end)
      const float* nq = (knext < DDIM) ? (bq + 4 * rs) : bq;
      knext += 4;
      const float nb0 = nq[0];
      const float nb1 = nq[rs];

      ss = fmaf(b0, b0, fmaf(b1, b1, ss));
      v2f bfrag;
      bfrag[0] = b0;
      bfrag[1] = b1;
      const int klocal = kk + 2 * half;  // even -> 8B-aligned ds_load_b64
#pragma unroll
      for (int mt = 0; mt < 4; ++mt) {
        const v2f afrag = *(const v2f*)&ldsA[(mt * 16 + ml) * LDSS + klocal];
#if __has_builtin(__builtin_amdgcn_wmma_f32_16x16x4_f32)
        acc[mt] = __builtin_amdgcn_wmma_f32_16x16x4_f32(
            false, afrag, false, bfrag, (short)0, acc[mt], false, false);
#else
#pragma unroll
        for (int r = 0; r < 8; ++r)
          acc[mt][r] = fmaf(afrag[0], bfrag[0], fmaf(afrag[1], bfrag[1], acc[mt][r]));
#endif
      }
      bq = nq;
      b0 = nb0;
      b1 = nb1;
    }
    __syncthreads();
  }

  // scatter D tiles: rows m < 50 only
#pragma unroll
  for (int mt = 0; mt < 4; ++mt) {
    const int mbase = mt * 16 + 8 * half;
#pragma unroll
    for (int r = 0; r < 8; ++r) {
      const int m = mbase + r;
      if (m < CCON && cok) outRows[(size_t)m * ncols + col] = acc[mt][r];
    }
  }

  if (tesq) {
    const float tot = ss + __shfl_xor(ss, 16, 32);
    if (half == 0 && cok) tesq[col] = tot;
  }
}

// ---------------------------------------------------------------------------
// W3 = (hx_weight @ concept) @ invG   -> (4 x 50)
// ---------------------------------------------------------------------------
__global__ __launch_bounds__(256) void w3_kernel(const float* __restrict__ hxw,
                                                 const float* __restrict__ con,
                                                 const float* __restrict__ invg,
                                                 float* __restrict__ w3) {
  __shared__ float w2[NCLS * CCON];
  const int tid = threadIdx.x;
  if (tid < NCLS * CCON) {
    const int m = tid / CCON, c = tid % CCON;
    float s = 0.f;
    for (int d = 0; d < DDIM; ++d)
      s = fmaf(hxw[m * DDIM + d], con[d * CCON + c], s);
    w2[tid] = s;
  }
  __syncthreads();
  if (tid < NCLS * CCON) {
    const int m = tid / CCON, c = tid % CCON;
    float s = 0.f;
    for (int j = 0; j < CCON; ++j)
      s = fmaf(w2[m * CCON + j], invg[j * CCON + c], s);
    w3[tid] = s;
  }
}

// ---------------------------------------------------------------------------
// orig_pred[b,m] = X[b,:]·hxW[m,:] + bias[m]
// y_pred[b,m]    = sum_c W3[m,c]*T1[c,b] + bias[m]
// one wave per batch row; lane-strided + shfl_xor reduction
// ---------------------------------------------------------------------------
__global__ __launch_bounds__(256) void pred_kernel(
    const float* __restrict__ X, const float* __restrict__ hxw,
    const float* __restrict__ bias, const float* __restrict__ T1,
    const float* __restrict__ w3, float* __restrict__ out) {
  const int tid = threadIdx.x, wave = tid >> 5, lane = tid & 31;
  const int b = blockIdx.x * 8 + wave;
  float accO[NCLS] = {0.f, 0.f, 0.f, 0.f};
  float accY[NCLS] = {0.f, 0.f, 0.f, 0.f};
  for (int d = lane; d < DDIM; d += 32) {
    const float x = X[(size_t)b * DDIM + d];
#pragma unroll
    for (int m = 0; m < NCLS; ++m) accO[m] = fmaf(x, hxw[m * DDIM + d], accO[m]);
  }
  for (int c = lane; c < CCON; c += 32) {
    const float t = T1[(size_t)c * BB + b];
#pragma unroll
    for (int m = 0; m < NCLS; ++m) accY[m] = fmaf(w3[m * CCON + c], t, accY[m]);
  }
#pragma unroll
  for (int off = 16; off; off >>= 1) {
#pragma unroll
    for (int m = 0; m < NCLS; ++m) {
      accO[m] += __shfl_xor(accO[m], off, 32);
      accY[m] += __shfl_xor(accY[m], off, 32);
    }
  }
  if (lane == 0) {
#pragma unroll
    for (int m = 0; m < NCLS; ++m) {
      out[(size_t)b * NCLS + m] = accO[m] + bias[m];
      out[(size_t)BB * NCLS + (size_t)b * NCLS + m] = accY[m] + bias[m];
    }
  }
}

// ---------------------------------------------------------------------------
// Per-concept top-k by rising-threshold min scans over key = te_sq - 2*dots.
// Accumulates the dot values of the k nearest neighbors -> csum[c].
// Deterministic (index tie-break; sequential accumulation of selections).
// ---------------------------------------------------------------------------
__global__ __launch_bounds__(256) void topk_kernel(const float* __restrict__ dots,
                                                   const float* __restrict__ tesq,
                                                   const int* __restrict__ kptr,
                                                   float* __restrict__ csum) {
  __shared__ float sv[256];
  __shared__ int si[256];
  __shared__ float sthresh, ssum;
  const int c = blockIdx.x, tid = threadIdx.x;
  const float* row = dots + (size_t)c * NTE;
  int k = *kptr;
  if (k < 0) k = 0;
  if (k > 1024) k = 1024;
  if (tid == 0) { sthresh = -FLT_MAX; ssum = 0.f; }
  __syncthreads();
  for (int it = 0; it < k; ++it) {
    const float thresh = sthresh;
    float best = FLT_MAX;
    int bidx = 0x7fffffff;
    for (int n = tid; n < NTE; n += 256) {
      const float key = fmaf(-2.0f, row[n], tesq[n]);
      if (key > thresh && (key < best || (key == best && n < bidx))) {
        best = key;
        bidx = n;
      }
    }
    sv[tid] = best;
    si[tid] = bidx;
    __syncthreads();
    for (int s = 128; s; s >>= 1) {
      if (tid < s) {
        const float ov = sv[tid + s];
        const int oi = si[tid + s];
        if (ov < sv[tid] || (ov == sv[tid] && oi < si[tid])) { sv[tid] = ov; si[tid] = oi; }
      }
      __syncthreads();
    }
    if (tid == 0) {
      sthresh = sv[0];
      if (si[0] < NTE) ssum += row[si[0]];
    }
    __syncthreads();
  }
  if (tid == 0) csum[c] = ssum;
}

// ---------------------------------------------------------------------------
// Final scalars: L_sparse_1, L_sparse_2, norm_metrics, similarity_penalty
// ---------------------------------------------------------------------------
__global__ __launch_bounds__(256) void scalars_kernel(const float* __restrict__ gram,
                                                      const float* __restrict__ csum,
                                                      const int* __restrict__ kptr,
                                                      float* __restrict__ out) {
  __shared__ float s_off[256], s_diag[256], s_abs[256];
  const int tid = threadIdx.x;
  float so = 0.f, sd = 0.f, sa = 0.f;
  for (int idx = tid; idx < CCON * CCON; idx += 256) {
    const int i = idx / CCON, j = idx % CCON;
    const float g = gram[idx];
    const float e = (i == j) ? 1.f : 0.f;
    if (i == j) sd += g; else so += g;
    sa += fabsf(g - e);
  }
  s_off[tid] = so; s_diag[tid] = sd; s_abs[tid] = sa;
  __syncthreads();
  for (int s = 128; s; s >>= 1) {
    if (tid < s) {
      s_off[tid] += s_off[tid + s];
      s_diag[tid] += s_diag[tid + s];
      s_abs[tid] += s_abs[tid + s];
    }
    __syncthreads();
  }
  if (tid == 0) {
    float l1 = 0.f;
    for (int cc = 0; cc < CCON; ++cc) l1 += csum[cc];
    int k = *kptr;
    if (k <= 0) k = 1;
    const float denom = (float)(CCON * CCON);
    out[2 * BB * NCLS + 0] = l1 / (float)(k * CCON);
    out[2 * BB * NCLS + 1] = s_off[0] / denom;
    out[2 * BB * NCLS + 2] = s_diag[0] / denom;
    out[2 * BB * NCLS + 3] = s_abs[0] / denom;
  }
}

// ---------------------------------------------------------------------------
extern "C" void kernel_launch(void* const* d_in, const int* in_sizes, int n_in,
                              void* d_out, int out_size, void* d_ws, size_t ws_size,
                              hipStream_t stream) {
  (void)in_sizes; (void)n_in; (void)out_size; (void)ws_size;
  const float* con  = (const float*)d_in[0];   // (D, C)
  const float* te   = (const float*)d_in[1];   // (D, N)
  const float* X    = (const float*)d_in[2];   // (B, D)
  const float* hxw  = (const float*)d_in[3];   // (4, D)
  const float* bias = (const float*)d_in[4];   // (4,)
  const int*   kptr = (const int*)d_in[5];     // scalar topk

  float* out  = (float*)d_out;
  float* ws   = (float*)d_ws;
  float* dots = ws + OFF_DOTS;
  float* tesq = ws + OFF_TESQ;
  float* xt   = ws + OFF_XT;
  float* t1   = ws + OFF_T1;
  float* gram = ws + OFF_GRAM;
  float* invg = ws + OFF_INVG;
  float* w3   = ws + OFF_W3;
  float* csum = ws + OFF_CSUM;
  float* apad = ws + OFF_APAD;

  padA_kernel<<<(MPAD * DDIM + 255) / 256, 256, 0, stream>>>(con, apad);
  transpose_kernel<<<dim3(DDIM / 32, BB / 32), dim3(32, 8), 0, stream>>>(X, xt);
  gram_kernel<<<(CCON * CCON + 255) / 256, 256, 0, stream>>>(con, gram);
  inv_kernel<<<1, 64, 0, stream>>>(gram, invg);
  // flagship: dots = concept^T @ TE, fused te_sq
  wmma_gemm_kernel<<<(NTE + 127) / 128, 256, 0, stream>>>(apad, te, dots, tesq, NTE);
  // T1 = concept^T @ X^T
  wmma_gemm_kernel<<<(BB + 127) / 128, 256, 0, stream>>>(apad, xt, t1, nullptr, BB);
  w3_kernel<<<1, 256, 0, stream>>>(hxw, con, invg, w3);
  pred_kernel<<<BB / 8, 256, 0, stream>>>(X, hxw, bias, t1, w3, out);
  topk_kernel<<<CCON, 256, 0, stream>>>(dots, tesq, kptr, csum);
  scalars_kernel<<<1, 256, 0, stream>>>(gram, csum, kptr, out);
}